// SpatialAttention_19765439496916
// MI455X (gfx1250) — compile-verified
//
#include <hip/hip_runtime.h>

typedef __attribute__((ext_vector_type(2))) float v2f;
typedef __attribute__((ext_vector_type(8))) float v8f;

#define Gg 2
#define Nn 192
#define Dd 64
#define Hh 4
#define DKk 16

__device__ __forceinline__ v8f wmma_f32(v2f a, v2f b, v8f c) {
  // D = A(16x4 f32) * B(4x16 f32) + C(16x16 f32)
  return __builtin_amdgcn_wmma_f32_16x16x4_f32(
      /*neg_a=*/false, a, /*neg_b=*/false, b,
      /*c_mod=*/(short)0, c, /*reuse_a=*/false, /*reuse_b=*/false);
}

// Y(16x16 tile) = X(Rx64) * W(64x64) + bias, one wave per tile.
// headScatter=1: write to (gn, h=colTile, m, dk) layout for q/k/v.
__global__ __launch_bounds__(256)
void proj16(const float* __restrict__ X, const float* __restrict__ W,
            const float* __restrict__ bias, float* __restrict__ Y,
            int numRowTiles, int headScatter)
{
  const int lane  = threadIdx.x & 31;
  const int wv    = threadIdx.x >> 5;
  const long tile = (long)blockIdx.x * 8 + wv;
  if (tile >= (long)numRowTiles * 4) return;   // wave-uniform

  const int  ct    = (int)(tile & 3);          // column tile == head
  const long rt    = tile >> 2;
  const int  n0    = ct * 16;
  const int  lhalf = lane >> 4;
  const int  l15   = lane & 15;
  const int  kb    = lhalf * 2;

  // Preload B fragments for all 16 K-steps (W is (in,out) row-major)
  v2f bfrag[16];
#pragma unroll
  for (int k = 0; k < 16; ++k) {
    const int kk = 4 * k + kb;
    bfrag[k].x = W[kk * 64 + n0 + l15];
    bfrag[k].y = W[(kk + 1) * 64 + n0 + l15];
  }

  // C initialized with bias broadcast down rows
  const float bb = bias[n0 + l15];
  v8f c;
#pragma unroll
  for (int r = 0; r < 8; ++r) c[r] = bb;

  const float* xrow = X + (rt * 16 + l15) * 64;
#pragma unroll
  for (int k = 0; k < 16; ++k) {
    v2f a = *(const v2f*)(xrow + 4 * k + kb);
    c = wmma_f32(a, bfrag[k], c);
  }

  const long row0 = rt * 16;
  if (headScatter) {
#pragma unroll
    for (int r = 0; r < 8; ++r) {
      const long rowg = row0 + r + lhalf * 8;
      const long gn   = rowg / Nn;
      const long m    = rowg % Nn;
      Y[((gn * Hh + ct) * Nn + m) * DKk + l15] = c[r];
    }
  } else {
#pragma unroll
    for (int r = 0; r < 8; ++r) {
      const long rowg = row0 + r + lhalf * 8;
      Y[rowg * 64 + n0 + l15] = c[r];
    }
  }
}

// One block (4 waves) per (g,n,h); each wave handles 3 strips of 16 rows.
__global__ __launch_bounds__(128)
void attn(const float* __restrict__ Q, const float* __restrict__ K,
          const float* __restrict__ V, float* __restrict__ Xout)
{
  __shared__ float lds[4][16 * Nn];   // 48 KB: per-wave P strip
  const int lane  = threadIdx.x & 31;
  const int wv    = threadIdx.x >> 5;
  const int gnh   = blockIdx.x;
  const int gn    = gnh / Hh;
  const int h     = gnh % Hh;
  const int lhalf = lane >> 4;
  const int l15   = lane & 15;
  const int kb    = lhalf * 2;

  const float* q = Q + (long)gnh * (Nn * DKk);
  const float* k = K + (long)gnh * (Nn * DKk);
  const float* v = V + (long)gnh * (Nn * DKk);
  float* myLds = lds[wv];

  for (int strip = wv; strip < 12; strip += 4) {  // uniform per wave
    const int m0 = strip * 16;

    // Q A-fragments (reused across 12 column tiles)
    v2f aq[4];
#pragma unroll
    for (int j = 0; j < 4; ++j)
      aq[j] = *(const v2f*)(q + (m0 + l15) * DKk + 4 * j + kb);

    // scores S = Q * K^T : 12 tiles x 4 WMMAs
    v8f sacc[12];
#pragma unroll
    for (int st = 0; st < 12; ++st) {
      v8f c;
#pragma unroll
      for (int r = 0; r < 8; ++r) c[r] = 0.f;
#pragma unroll
      for (int j = 0; j < 4; ++j) {
        v2f bf = *(const v2f*)(k + (st * 16 + l15) * DKk + 4 * j + kb);
        c = wmma_f32(aq[j], bf, c);
      }
      sacc[st] = c;
    }

    // softmax over each row (per accumulator component r)
#pragma unroll
    for (int r = 0; r < 8; ++r) {
      float rm = -3.402823466e38f;
#pragma unroll
      for (int st = 0; st < 12; ++st) {
        sacc[st][r] *= 0.25f;                  // 1/sqrt(DK)
        rm = fmaxf(rm, sacc[st][r]);
      }
#pragma unroll
      for (int off = 1; off < 16; off <<= 1)
        rm = fmaxf(rm, __shfl_xor(rm, off, 32));
      float rs = 0.f;
#pragma unroll
      for (int st = 0; st < 12; ++st) {
        const float p = __expf(sacc[st][r] - rm);
        sacc[st][r] = p;
        rs += p;
      }
#pragma unroll
      for (int off = 1; off < 16; off <<= 1)
        rs += __shfl_xor(rs, off, 32);
      const float inv = 1.0f / rs;
#pragma unroll
      for (int st = 0; st < 12; ++st) sacc[st][r] *= inv;
    }

    // spill P (16 x 192, row major) through this wave's LDS slice
#pragma unroll
    for (int st = 0; st < 12; ++st)
#pragma unroll
      for (int r = 0; r < 8; ++r)
        myLds[(r + lhalf * 8) * Nn + st * 16 + l15] = sacc[st][r];

    // X = P * V : 12 chunks x 4 WMMAs, A re-read from LDS in A-frag layout
    v8f xc;
#pragma unroll
    for (int r = 0; r < 8; ++r) xc[r] = 0.f;
#pragma unroll
    for (int sc = 0; sc < 12; ++sc) {
#pragma unroll
      for (int j = 0; j < 4; ++j) {
        v2f ap = *(const v2f*)(myLds + l15 * Nn + sc * 16 + 4 * j + kb);
        const int ss = sc * 16 + 4 * j + kb;
        v2f bv;
        bv.x = v[ss * DKk + l15];
        bv.y = v[(ss + 1) * DKk + l15];
        xc = wmma_f32(ap, bv, xc);
      }
    }

    // store X tile into (g,n,m, h*16+dk) layout
#pragma unroll
    for (int r = 0; r < 8; ++r) {
      const long rowflat = (long)gn * Nn + m0 + r + lhalf * 8;
      Xout[rowflat * 64 + h * 16 + l15] = xc[r];
    }
  }
}

extern "C" void kernel_launch(void* const* d_in, const int* in_sizes, int n_in,
                              void* d_out, int out_size, void* d_ws, size_t ws_size,
                              hipStream_t stream) {
  const float* adj = (const float*)d_in[0];
  const float* Wq  = (const float*)d_in[1];
  const float* bq  = (const float*)d_in[2];
  const float* Wk  = (const float*)d_in[3];
  const float* bk  = (const float*)d_in[4];
  const float* Wv  = (const float*)d_in[5];
  const float* bv  = (const float*)d_in[6];
  const float* Wo  = (const float*)d_in[7];
  const float* bo  = (const float*)d_in[8];
  float* out = (float*)d_out;

  const long R        = (long)Gg * Nn * Nn;   // 73728 flattened rows
  const long bufElems = R * Dd;               // 4.72M floats per buffer
  float* qb = (float*)d_ws;
  float* kb = qb + bufElems;
  float* vb = kb + bufElems;
  float* xb = vb + bufElems;

  const int numRowTiles = (int)(R / 16);      // 4608
  const int tiles       = numRowTiles * 4;    // 18432 (8 waves / block)
  const int blocks      = (tiles + 7) / 8;    // 2304

  proj16<<<blocks, 256, 0, stream>>>(adj, Wq, bq, qb, numRowTiles, 1);
  proj16<<<blocks, 256, 0, stream>>>(adj, Wk, bk, kb, numRowTiles, 1);
  proj16<<<blocks, 256, 0, stream>>>(adj, Wv, bv, vb, numRowTiles, 1);
  attn<<<Gg * Nn * Hh, 128, 0, stream>>>(qb, kb, vb, xb);
  proj16<<<blocks, 256, 0, stream>>>(xb, Wo, bo, out, numRowTiles, 0);
}